// LinAttnManual_21208548507759
// MI455X (gfx1250) — compile-verified
//
#include <hip/hip_runtime.h>

// ---------------------------------------------------------------------------
// Taylor-series linear attention == polynomial-kernel causal attention:
//   score(i,j) = 1 + (q.k)/4 + (q.k)^2/32   for j <= i
//   out_i = sum_j score*v_j / sum_j score
// FP32 WMMA (V_WMMA_F32_16X16X4_F32) + async global->LDS tile staging.
// ---------------------------------------------------------------------------

typedef float v2f __attribute__((ext_vector_type(2)));
typedef float v4fl __attribute__((ext_vector_type(4)));
typedef float v8f __attribute__((ext_vector_type(8)));
typedef int   i4v __attribute__((vector_size(16)));  // matches async builtin

static constexpr int BATCH    = 4;
static constexpr int SEQ      = 2048;
static constexpr int DM       = 32;
static constexpr int NH       = 2;
static constexpr int FD       = 16;
static constexpr int BH       = BATCH * NH;        // 8 heads total
static constexpr int QTILES   = SEQ / 16;          // 128 query tiles per head
static constexpr int ROWTILES = BATCH * SEQ / 16;  // 512 row tiles of X
static constexpr int WAVES    = 4;                 // waves per block

static constexpr int KVLD = 20;  // padded LDS row pitch (floats) for K/V tiles
                                 // 20*4=80B keeps 16B alignment for b128 async
                                 // writes and makes ds fragment reads
                                 // bank-conflict-free (n*20+k distinct mod 64)
static constexpr int PLD  = 17;  // padded pitch for the P relayout tile

#define WMMA_F32X4(a, b, c) \
  __builtin_amdgcn_wmma_f32_16x16x4_f32(false, (a), false, (b), (short)0, (c), false, false)

// ---- async global->LDS staging (CDNA5 ASYNCcnt path), with safe fallback ---
#if defined(__has_builtin)
#if __has_builtin(__builtin_amdgcn_global_load_async_to_lds_b128)
#define USE_ASYNC_LDS 1
#endif
#endif

#ifdef USE_ASYNC_LDS
#if __has_builtin(__builtin_amdgcn_s_wait_asynccnt)
#define WAIT_ASYNC(n)                         \
  do {                                        \
    __builtin_amdgcn_s_wait_asynccnt(n);      \
    asm volatile("" ::: "memory");            \
  } while (0)
#else
#define WAIT_ASYNC(n) asm volatile("s_wait_asynccnt %0" ::"i"(n) : "memory")
#endif
#else
#define WAIT_ASYNC(n) asm volatile("" ::: "memory")
#endif

typedef __attribute__((address_space(1))) i4v gi4;  // global 16B vector
typedef __attribute__((address_space(3))) i4v li4;  // LDS 16B vector

// Copy one 16x16 f32 tile (row-major, 64B rows) from global into an LDS tile
// with row pitch KVLD floats. Each lane moves two 16-byte chunks.
__device__ __forceinline__ void stage_tile(const float* gsrc, float* ldst, int lane) {
  int r  = lane >> 2;  // row 0..7 (and +8 for second chunk)
  int cq = lane & 3;   // 16B chunk within the 64B row
#ifdef USE_ASYNC_LDS
  gi4* g = (gi4*)gsrc;  // 16B units: 4 per 64B global row
  li4* l = (li4*)ldst;  // 16B units: KVLD/4 = 5 per padded LDS row
  __builtin_amdgcn_global_load_async_to_lds_b128(
      g + (r * 4 + cq), l + (r * (KVLD / 4) + cq), 0, 0);
  __builtin_amdgcn_global_load_async_to_lds_b128(
      g + ((r + 8) * 4 + cq), l + ((r + 8) * (KVLD / 4) + cq), 0, 0);
#else
  *(v4fl*)(ldst + r * KVLD + cq * 4) = *(const v4fl*)(gsrc + r * 16 + cq * 4);
  *(v4fl*)(ldst + (r + 8) * KVLD + cq * 4) =
      *(const v4fl*)(gsrc + (r + 8) * 16 + cq * 4);
#endif
}

// A-fragment (16x4 f32) chunk c from a row-major [*, ld] tile.
// lanes 0-15: M=lane, K=4c+{0,1}; lanes 16-31: M=lane-16, K=4c+{2,3}.
__device__ __forceinline__ v2f load_a_chunk(const float* src, int ld, int c, int lane) {
  int row = lane & 15;
  int k   = 4 * c + 2 * (lane >> 4);
  v2f a;
  a.x = src[row * ld + k];
  a.y = src[row * ld + k + 1];
  return a;
}

// B-fragment (4x16 f32) chunk c where B[k][n] = src[k*ld + n].
__device__ __forceinline__ v2f load_b_chunk(const float* src, int ld, int c, int lane) {
  int n = lane & 15;
  int k = 4 * c + 2 * (lane >> 4);
  v2f b;
  b.x = src[k * ld + n];
  b.y = src[(k + 1) * ld + n];
  return b;
}

// B-fragment of K^T: B[k][n] = K[n][k] with K row-major [*, ld].
__device__ __forceinline__ v2f load_bT_chunk(const float* src, int ld, int c, int lane) {
  int n = lane & 15;
  int k = 4 * c + 2 * (lane >> 4);
  v2f b;
  b.x = src[n * ld + k];
  b.y = src[n * ld + k + 1];
  return b;
}

// C/D fragment (16x16 f32, 8 VGPRs) store to row-major [*, ld] tile.
__device__ __forceinline__ void store_c_tile(float* dst, int ld, v8f c, int lane) {
  int col = lane & 15;
  int mo  = (lane >> 4) * 8;
#pragma unroll
  for (int r = 0; r < 8; ++r) dst[(mo + r) * ld + col] = c[r];
}

// ---------------------------------------------------------------------------
// Kernel 1: QKV projection. X[B*L,32] @ W[32,32] -> per-head [BH, L, 16].
// ---------------------------------------------------------------------------
__global__ __launch_bounds__(WAVES * 32) void lin_attn_qkv_proj(
    const float* __restrict__ X,
    const float* __restrict__ Wq, const float* __restrict__ Wk,
    const float* __restrict__ Wv,
    float* __restrict__ Qh, float* __restrict__ Kh, float* __restrict__ Vh) {
  int lane = threadIdx.x & 31;
  int wid  = threadIdx.x >> 5;
  int t    = blockIdx.x * WAVES + wid;  // row tile index [0, 512)
  int b    = (t * 16) / SEQ;
  int l0   = (t * 16) % SEQ;

  const float* Xt = X + (size_t)t * 16 * DM;
  v2f xa[8];
#pragma unroll
  for (int c = 0; c < 8; ++c) xa[c] = load_a_chunk(Xt, DM, c, lane);

  const float* Ws[3]  = {Wq, Wk, Wv};
  float*       Out[3] = {Qh, Kh, Vh};

#pragma unroll
  for (int w = 0; w < 3; ++w) {
#pragma unroll
    for (int h = 0; h < NH; ++h) {
      v8f acc = {};
#pragma unroll
      for (int c = 0; c < 8; ++c) {
        v2f bf = load_b_chunk(Ws[w] + h * FD, DM, c, lane);
        acc = WMMA_F32X4(xa[c], bf, acc);
      }
      float* dst = Out[w] + ((size_t)(b * NH + h) * SEQ + l0) * FD;
      store_c_tile(dst, FD, acc, lane);
    }
  }
}

// ---------------------------------------------------------------------------
// Kernel 2: causal polynomial attention. One wave per (head, query tile).
// K/V tiles double-buffered in LDS via async global->LDS DMA.
// ---------------------------------------------------------------------------
__global__ __launch_bounds__(WAVES * 32) void lin_attn_poly_attn(
    const float* __restrict__ Qh, const float* __restrict__ Kh,
    const float* __restrict__ Vh, float* __restrict__ Yh) {
  __shared__ float Pl[WAVES][16 * PLD];        // C-frag -> A-frag relayout
  __shared__ float Kbuf[WAVES][2][16 * KVLD];  // double-buffered K tile
  __shared__ float Vbuf[WAVES][2][16 * KVLD];  // double-buffered V tile

  int lane = threadIdx.x & 31;
  int wid  = threadIdx.x >> 5;
  int gid  = blockIdx.x * WAVES + wid;  // [0, BH*QTILES)
  int bh   = gid / QTILES;
  int qi   = gid % QTILES;

  const float* Kbase = Kh + (size_t)bh * SEQ * FD;
  const float* Vbase = Vh + (size_t)bh * SEQ * FD;

  const float* Qt = Qh + ((size_t)bh * SEQ + qi * 16) * FD;
  v2f qa[4];
#pragma unroll
  for (int c = 0; c < 4; ++c) qa[c] = load_a_chunk(Qt, FD, c, lane);

  v8f nacc  = {};           // numerator [16 x 16]
  v8f dacc  = {};           // denominator (all columns = row-sum of P)
  v2f onesb = {1.0f, 1.0f};

  int col = lane & 15;        // N index (key within tile / feature)
  int mo  = (lane >> 4) * 8;  // M offset of this lane's half

  // Prologue: stage tile j=0.
  stage_tile(Kbase, &Kbuf[wid][0][0], lane);
  stage_tile(Vbase, &Vbuf[wid][0][0], lane);

  for (int j = 0; j <= qi; ++j) {
    int cur = j & 1;
    if (j < qi) {
      // WAR: make sure prior ds reads of the other buffer retired before the
      // async engine overwrites it (compiler can't see through the DMA).
      asm volatile("s_wait_dscnt 0x0" ::: "memory");
      stage_tile(Kbase + (size_t)(j + 1) * 16 * FD, &Kbuf[wid][cur ^ 1][0], lane);
      stage_tile(Vbase + (size_t)(j + 1) * 16 * FD, &Vbuf[wid][cur ^ 1][0], lane);
      WAIT_ASYNC(4);  // 8 outstanding -> first 4 (current tile) complete
    } else {
      WAIT_ASYNC(0);
    }
    const float* Kt = &Kbuf[wid][cur][0];
    const float* Vt = &Vbuf[wid][cur][0];

    // S = Q @ K^T (16x16, k=16 in four 4-chunks)
    v8f s = {};
#pragma unroll
    for (int c = 0; c < 4; ++c) {
      v2f kb = load_bT_chunk(Kt, KVLD, c, lane);
      s = WMMA_F32X4(qa[c], kb, s);
    }

    // P = 1 + S/4 + S^2/32, causal mask on the diagonal tile (keep k <= m)
    v8f p;
#pragma unroll
    for (int r = 0; r < 8; ++r) {
      float u    = s[r];
      float val  = fmaf(u, fmaf(u, 0.03125f, 0.25f), 1.0f);
      bool  dead = (j == qi) && (col > mo + r);
      p[r] = dead ? 0.0f : val;
    }

    // Stage P through LDS (row-major, padded) to get A-fragment layout.
#pragma unroll
    for (int r = 0; r < 8; ++r) Pl[wid][(mo + r) * PLD + col] = p[r];

#pragma unroll
    for (int c = 0; c < 4; ++c) {
      v2f pa = load_a_chunk(&Pl[wid][0], PLD, c, lane);
      v2f vb = load_b_chunk(Vt, KVLD, c, lane);
      nacc = WMMA_F32X4(pa, vb, nacc);     // numerator += P @ V
      dacc = WMMA_F32X4(pa, onesb, dacc);  // denom(row-sums) += P @ ones
    }
  }

  // dacc has the row-sum replicated in every column -> same lane pairing.
  float* Yt = Yh + ((size_t)bh * SEQ + qi * 16) * FD;
#pragma unroll
  for (int r = 0; r < 8; ++r) Yt[(mo + r) * FD + col] = nacc[r] / dacc[r];
}

// ---------------------------------------------------------------------------
// Kernel 3: output projection. Y[B*L,32] (gathered from heads) @ Wo[32,32].
// ---------------------------------------------------------------------------
__global__ __launch_bounds__(WAVES * 32) void lin_attn_out_proj(
    const float* __restrict__ Yh, const float* __restrict__ Wo,
    float* __restrict__ Out) {
  int lane = threadIdx.x & 31;
  int wid  = threadIdx.x >> 5;
  int t    = blockIdx.x * WAVES + wid;  // row tile [0, 512)
  int b    = (t * 16) / SEQ;
  int l0   = (t * 16) % SEQ;

  v2f ya[8];
#pragma unroll
  for (int c = 0; c < 8; ++c) {
    int row  = lane & 15;
    int k    = 4 * c + 2 * (lane >> 4);  // column of assembled Y in [0,32)
    int head = k >> 4;
    int f    = k & 15;
    const float* src =
        Yh + ((size_t)(b * NH + head) * SEQ + l0 + row) * FD + f;
    v2f a;
    a.x = src[0];
    a.y = src[1];  // chunks never straddle the head boundary (f <= 14 here)
    ya[c] = a;
  }

#pragma unroll
  for (int n0 = 0; n0 < DM; n0 += 16) {
    v8f acc = {};
#pragma unroll
    for (int c = 0; c < 8; ++c) {
      v2f bf = load_b_chunk(Wo + n0, DM, c, lane);
      acc = WMMA_F32X4(ya[c], bf, acc);
    }
    store_c_tile(Out + (size_t)t * 16 * DM + n0, DM, acc, lane);
  }
}

// ---------------------------------------------------------------------------
extern "C" void kernel_launch(void* const* d_in, const int* in_sizes, int n_in,
                              void* d_out, int out_size, void* d_ws, size_t ws_size,
                              hipStream_t stream) {
  const float* X  = (const float*)d_in[0];
  const float* Wq = (const float*)d_in[1];
  const float* Wk = (const float*)d_in[2];
  const float* Wv = (const float*)d_in[3];
  const float* Wo = (const float*)d_in[4];

  float* ws  = (float*)d_ws;
  size_t per = (size_t)BH * SEQ * FD;  // 262144 floats = 1 MB
  float* Qh  = ws;
  float* Kh  = ws + per;
  float* Vh  = ws + 2 * per;
  float* Yh  = ws + 3 * per;

  lin_attn_qkv_proj<<<ROWTILES / WAVES, WAVES * 32, 0, stream>>>(
      X, Wq, Wk, Wv, Qh, Kh, Vh);
  lin_attn_poly_attn<<<(BH * QTILES) / WAVES, WAVES * 32, 0, stream>>>(
      Qh, Kh, Vh, Yh);
  lin_attn_out_proj<<<ROWTILES / WAVES, WAVES * 32, 0, stream>>>(
      Yh, Wo, (float*)d_out);
}